// Rnn_net_22754736734783
// MI455X (gfx1250) — compile-verified
//
#include <hip/hip_runtime.h>
#include <hip/hip_bf16.h>

// 2-layer tanh RNN (H=10) + per-step Linear(10->1), B=4096, T=2048.
// Latency-bound sequential recurrence -> keep all weights/biases/state in
// registers; per-timestep math = 9x V_WMMA_F32_16X16X4_F32 per 16-batch tile.
// Layout: hidden state kept transposed D[m=hidden,n=batch] in WMMA C/D layout;
// weights pre-packed into WMMA A layout (16x12, zero padded, 3 K4 chunks).
// State relayout (D -> B operand) uses v_permlanex16_b32 (pure VALU lane^16
// exchange) instead of ds_bpermute to keep LDS waits off the serial chain.

typedef __attribute__((ext_vector_type(2))) float v2f;
typedef __attribute__((ext_vector_type(8))) float v8f;

#if __has_builtin(__builtin_amdgcn_tanhf)
#define TANHF(v) __builtin_amdgcn_tanhf(v)
#elif __has_builtin(__builtin_amdgcn_tanh_f32)
#define TANHF(v) __builtin_amdgcn_tanh_f32(v)
#else
#define TANHF(v) tanhf(v)
#endif

// lane <-> lane^16 exchange (wave32): v_permlanex16_b32 with identity selects.
static __device__ __forceinline__ float xor16(float v) {
#if __has_builtin(__builtin_amdgcn_permlanex16)
    unsigned u = __builtin_bit_cast(unsigned, v);
    u = __builtin_amdgcn_permlanex16(u, u, 0x76543210u, 0xFEDCBA98u,
                                     /*fi=*/false, /*bound_ctrl=*/false);
    return __builtin_bit_cast(float, u);
#else
    return __shfl_xor(v, 16);
#endif
}

static __device__ __forceinline__ v8f wmma4(v2f a, v2f b, v8f c) {
    // D = A(16x4,f32) x B(4x16,f32) + C(16x16,f32)
    return __builtin_amdgcn_wmma_f32_16x16x4_f32(
        /*neg_a=*/false, a, /*neg_b=*/false, b,
        /*c_mod=*/(short)0, c, /*reuse_a=*/false, /*reuse_b=*/false);
}

// Convert D-layout state (VGPR r: lanes0-15 = row r, lanes16-31 = row r+8)
// into three B-layout K4 chunks (chunk c, comp v: K = 4c+v (+2 in hi lanes)).
// Rows 10..11 of the K=12 padding are forced to zero.
static __device__ __forceinline__ void d_to_b(const v8f d, v2f b[3], bool hi) {
    float t0 = xor16(d[0]);
    float t1 = xor16(d[1]);
    float t2 = xor16(d[2]);
    float t3 = xor16(d[3]);
    float t6 = xor16(d[6]);
    float t7 = xor16(d[7]);
    b[0].x = hi ? t2 : (float)d[0];  b[0].y = hi ? t3 : (float)d[1];  // K 0,1 | 2,3
    b[1].x = hi ? t6 : (float)d[4];  b[1].y = hi ? t7 : (float)d[5];  // K 4,5 | 6,7
    b[2].x = hi ? 0.f : t0;          b[2].y = hi ? 0.f : t1;          // K 8,9 | 10,11(=0)
}

__global__ __launch_bounds__(256) void rnn2_wmma_kernel(
    const float* __restrict__ x,     // [B,T]
    const float* __restrict__ hs,    // [2,B,10]
    const float* __restrict__ Wih0,  // [10,1]
    const float* __restrict__ Whh0,  // [10,10]
    const float* __restrict__ bih0, const float* __restrict__ bhh0,
    const float* __restrict__ Wih1,  // [10,10]
    const float* __restrict__ Whh1,  // [10,10]
    const float* __restrict__ bih1, const float* __restrict__ bhh1,
    const float* __restrict__ Wout,  // [1,10]
    const float* __restrict__ bout,  // [1]
    float* __restrict__ outs,        // [B,T]
    float* __restrict__ hout,        // [2,B,10]
    int B, int T)
{
    const int lane = threadIdx.x & 31;
    const int wave = (int)((blockIdx.x * blockDim.x + threadIdx.x) >> 5);
    const int b0   = wave * 16;
    if (b0 >= B) return;                 // uniform per-wave: EXEC stays all-1s
    const int  n  = lane & 15;           // batch column within tile
    const bool hi = lane >= 16;
    const int  m  = n;                   // output-row index for A operand

    // ---- pre-pack weights into WMMA A layout (3 chunks of K=4, zero padded)
    v2f aHH0[3], aIH1[3], aHH1[3];
#pragma unroll
    for (int c = 0; c < 3; ++c) {
        int k0 = 4 * c + (hi ? 2 : 0);
        float w00 = (m < 10 && k0     < 10) ? Whh0[m * 10 + k0]     : 0.f;
        float w01 = (m < 10 && k0 + 1 < 10) ? Whh0[m * 10 + k0 + 1] : 0.f;
        float w10 = (m < 10 && k0     < 10) ? Wih1[m * 10 + k0]     : 0.f;
        float w11 = (m < 10 && k0 + 1 < 10) ? Wih1[m * 10 + k0 + 1] : 0.f;
        float w20 = (m < 10 && k0     < 10) ? Whh1[m * 10 + k0]     : 0.f;
        float w21 = (m < 10 && k0 + 1 < 10) ? Whh1[m * 10 + k0 + 1] : 0.f;
        aHH0[c].x = w00; aHH0[c].y = w01;
        aIH1[c].x = w10; aIH1[c].y = w11;
        aHH1[c].x = w20; aHH1[c].y = w21;
    }

    // ---- biases (folded into WMMA C start), input weight column, output row
    v8f cb0, cb1, wx, wo;
#pragma unroll
    for (int r = 0; r < 8; ++r) {
        int mm = hi ? r + 8 : r;
        cb0[r] = (mm < 10) ? (bih0[mm] + bhh0[mm]) : 0.f;
        cb1[r] = (mm < 10) ? (bih1[mm] + bhh1[mm]) : 0.f;
        wx[r]  = (mm < 10) ? Wih0[mm] : 0.f;
        wo[r]  = hi ? ((r < 2) ? Wout[8 + r] : 0.f) : Wout[r];
    }
    const float bo = bout[0];

    // ---- initial hidden state, gathered directly into B-chunk layout
    v2f bh0[3], bh1[3];
#pragma unroll
    for (int c = 0; c < 3; ++c) {
        int ka = 4 * c + (hi ? 2 : 0);
        int kb = ka + 1;
        const float* h0p = hs + (size_t)(b0 + n) * 10;
        const float* h1p = hs + (size_t)B * 10 + (size_t)(b0 + n) * 10;
        bh0[c].x = (ka < 10) ? h0p[ka] : 0.f;
        bh0[c].y = (kb < 10) ? h0p[kb] : 0.f;
        bh1[c].x = (ka < 10) ? h1p[ka] : 0.f;
        bh1[c].y = (kb < 10) ? h1p[kb] : 0.f;
    }

    v8f h0d = {}, h1d = {};
    const size_t xrow = (size_t)(b0 + n) * (size_t)T;

    for (int t = 0; t < T; ++t) {
        float xv = x[xrow + t];
        __builtin_prefetch(&x[xrow + t + 64], 0, 3);   // warm near caches

        // ---- layer 0: pre = Whh0 @ h0 + (bih0+bhh0) + x_t * Wih0 ; tanh
        v8f acc = wmma4(aHH0[0], bh0[0], cb0);
        acc = wmma4(aHH0[1], bh0[1], acc);
        acc = wmma4(aHH0[2], bh0[2], acc);
#pragma unroll
        for (int r = 0; r < 8; ++r) acc[r] = acc[r] + wx[r] * xv;
#pragma unroll
        for (int r = 0; r < 8; ++r) acc[r] = TANHF(acc[r]);
        h0d = acc;
        d_to_b(acc, bh0, hi);

        // ---- layer 1: pre = Wih1 @ h0_new + Whh1 @ h1 + biases ; tanh
        v8f a1 = wmma4(aIH1[0], bh0[0], cb1);
        a1 = wmma4(aIH1[1], bh0[1], a1);
        a1 = wmma4(aIH1[2], bh0[2], a1);
        a1 = wmma4(aHH1[0], bh1[0], a1);
        a1 = wmma4(aHH1[1], bh1[1], a1);
        a1 = wmma4(aHH1[2], bh1[2], a1);
#pragma unroll
        for (int r = 0; r < 8; ++r) a1[r] = TANHF(a1[r]);
        h1d = a1;
        d_to_b(a1, bh1, hi);

        // ---- out[b,t] = Wout . h1 + bout  (xor-16 half-lane reduction)
        float p = 0.f;
#pragma unroll
        for (int r = 0; r < 8; ++r) p += wo[r] * a1[r];
        p += xor16(p);
        if (!hi) outs[xrow + t] = p + bo;
    }

    // ---- final hidden state [2,B,10]
    const size_t hb = (size_t)(b0 + n) * 10;
    if (!hi) {
#pragma unroll
        for (int r = 0; r < 8; ++r) {
            hout[hb + r]                  = h0d[r];
            hout[(size_t)B * 10 + hb + r] = h1d[r];
        }
    } else {
#pragma unroll
        for (int r = 0; r < 2; ++r) {
            hout[hb + 8 + r]                  = h0d[r];
            hout[(size_t)B * 10 + hb + 8 + r] = h1d[r];
        }
    }
}

extern "C" void kernel_launch(void* const* d_in, const int* in_sizes, int n_in,
                              void* d_out, int out_size, void* d_ws, size_t ws_size,
                              hipStream_t stream) {
    const float* x    = (const float*)d_in[0];
    const float* hs   = (const float*)d_in[1];
    const float* Wih0 = (const float*)d_in[2];
    const float* Whh0 = (const float*)d_in[3];
    const float* bih0 = (const float*)d_in[4];
    const float* bhh0 = (const float*)d_in[5];
    const float* Wih1 = (const float*)d_in[6];
    const float* Whh1 = (const float*)d_in[7];
    const float* bih1 = (const float*)d_in[8];
    const float* bhh1 = (const float*)d_in[9];
    const float* Wout = (const float*)d_in[10];
    const float* bout = (const float*)d_in[11];

    const int B = in_sizes[1] / 20;          // hidden_state = [2,B,10]
    const int T = in_sizes[0] / B;           // x = [B,T,1]

    float* outs = (float*)d_out;             // [B,T]
    float* hout = outs + (size_t)B * T;      // [2,B,10]

    const int waves   = (B + 15) / 16;       // one wave per 16-batch tile
    const int threads = 256;                 // 8 waves/block (wave32)
    const int blocks  = (waves * 32 + threads - 1) / threads;

    rnn2_wmma_kernel<<<blocks, threads, 0, stream>>>(
        x, hs, Wih0, Whh0, bih0, bhh0, Wih1, Whh1, bih1, bhh1,
        Wout, bout, outs, hout, B, T);
}